// IPA_block_70841190580556
// MI455X (gfx1250) — compile-verified
//
#include <hip/hip_runtime.h>
#include <hip/hip_bf16.h>

// IPA block for MI455X (gfx1250, wave32). Bandwidth-bound on z (302 MB, > L2).
// All GEMMs use V_WMMA_F32_16X16X4_F32. Edge tiles handled branch-free via
// clamped addresses + 0/1 float masks; 32-bit offsets + uniform bases so the
// backend emits saddr global loads; K unrolled by 32 (8 chained WMMAs) for
// deep memory-level parallelism.

#define NRES 768
#define CS   384
#define CZ   128
#define CH   16
#define NH   12
#define PQ   4
#define PV   8

typedef float v2f __attribute__((ext_vector_type(2)));
typedef float v4f __attribute__((ext_vector_type(4)));
typedef float v8f __attribute__((ext_vector_type(8)));

// ---------------------------------------------------------------------------
// Generic batched WMMA f32 GEMM:  C[b] = A[b] (MxK) * B[b] (KxNc) + bias
// One wave computes one 16x16 C tile; K unrolled by 32 (8 chained WMMAs).
// A-matrix layout (ISA 7.12.2): lane l holds row (l&15), K-pair at
// k + ((l>>4)*2).  B/C/D row-striped across lanes.
// Requires K % 32 == 0 (all call sites: 384, 128, 768, 2208), even lda,
// and all flat offsets < 2^31 (max here ~75.5M).
// ---------------------------------------------------------------------------
__global__ __launch_bounds__(256)
void gemm_f32_wmma(const float* __restrict__ A, int lda, long Astride,
                   const float* __restrict__ Bm, int ldb, long Bstride,
                   const float* __restrict__ bias,
                   float* __restrict__ C, int ldc, long Cstride,
                   int M, int K, int Nc, int relu)
{
  const int lane  = threadIdx.x & 31;
  const int wave  = threadIdx.x >> 5;
  const int tilesN = (Nc + 15) >> 4;
  const int tilesM = (M  + 15) >> 4;
  const int tile = blockIdx.x * 8 + wave;          // wave-uniform
  if (tile >= tilesM * tilesN) return;             // uniform exit: EXEC all-1s
  const int tm = tile / tilesN;
  const int tn = tile % tilesN;

  A  += (size_t)blockIdx.y * (size_t)Astride;      // uniform base pointers
  Bm += (size_t)blockIdx.y * (size_t)Bstride;
  C  += (size_t)blockIdx.y * (size_t)Cstride;

  const int row  = tm * 16 + (lane & 15);
  const int col  = tn * 16 + (lane & 15);
  const int ksub = (lane >> 4) << 1;               // 0 or 2
  // branch-free edge handling: clamp address, zero via mask multiply
  const float am = (row < M)  ? 1.f : 0.f;
  const float bm = (col < Nc) ? 1.f : 0.f;
  const int rowc = row < M  ? row : (M - 1);
  const int colc = col < Nc ? col : (Nc - 1);

  // 32-bit per-lane offsets (uniform base + voffset -> saddr loads)
  unsigned aoff = (unsigned)rowc * (unsigned)lda + (unsigned)ksub;   // even
  unsigned boff = (unsigned)colc + (unsigned)ksub * (unsigned)ldb;
  // uniform B row-offset constants: {0,4,8,...,28} * ldb  (SGPRs)
  unsigned kldb[8];
#pragma unroll
  for (int i = 0; i < 8; ++i) kldb[i] = (unsigned)(4 * i) * (unsigned)ldb;
  const unsigned ldbu   = (unsigned)ldb;
  const unsigned bstep  = 32u * ldbu;

  v8f acc = {};
  for (int k = 0; k < K; k += 32) {
    v2f a[8], b[8];
#pragma unroll
    for (int s = 0; s < 8; ++s)                    // 8 aligned b64 loads
      a[s] = *(const v2f*)(A + aoff + 4u * (unsigned)s);
#pragma unroll
    for (int s = 0; s < 8; ++s) {                  // 16 lane-coalesced b32 loads
      b[s].x = Bm[boff + kldb[s]];
      b[s].y = Bm[boff + kldb[s] + ldbu];
    }
#pragma unroll
    for (int s = 0; s < 8; ++s) { a[s] *= am; b[s] *= bm; }
#pragma unroll
    for (int s = 0; s < 8; ++s)
      acc = __builtin_amdgcn_wmma_f32_16x16x4_f32(false, a[s], false, b[s],
                                                  (short)0, acc, false, false);
    aoff += 32u;
    boff += bstep;
  }
  // C/D layout: VGPR r -> row r (lanes 0-15) / row r+8 (lanes 16-31), N = lane&15
  const int mbase = tm * 16 + ((lane >> 4) << 3);
  const float badd = bias ? bias[colc] : 0.f;
#pragma unroll
  for (int r = 0; r < 8; ++r) {
    const int m = mbase + r;
    if (m < M && (col < Nc)) {
      float v = acc[r] + badd;
      if (relu) v = fmaxf(v, 0.f);
      C[(unsigned)m * (unsigned)ldc + (unsigned)col] = v;
    }
  }
}

// ---------------------------------------------------------------------------
// pts_global[n,p,:] = rot[n] @ pts_local[n,p,:] + trans[n]
// ---------------------------------------------------------------------------
__global__ void pts_to_global(const float* __restrict__ lin,
                              const float* __restrict__ rot,
                              const float* __restrict__ trans,
                              float* __restrict__ out, int npts)
{
  int idx = blockIdx.x * blockDim.x + threadIdx.x;
  if (idx >= NRES * npts) return;
  int n = idx / npts, p = idx % npts;
  const float* R  = rot + (size_t)n * 9;
  const float* in = lin + ((size_t)n * npts + p) * 3;
  float px = in[0], py = in[1], pz = in[2];
  float* o = out + ((size_t)n * npts + p) * 3;
  o[0] = R[0]*px + R[1]*py + R[2]*pz + trans[n*3+0];
  o[1] = R[3]*px + R[4]*py + R[5]*pz + trans[n*3+1];
  o[2] = R[6]*px + R[7]*py + R[8]*pz + trans[n*3+2];
}

// ---------------------------------------------------------------------------
// Per-(head, i-row) fused logits + softmax.  Full 768-wide row lives in LDS.
// ---------------------------------------------------------------------------
__global__ __launch_bounds__(256)
void attn_logits_softmax(const float* __restrict__ q,   const float* __restrict__ kv,
                         const float* __restrict__ qpts,const float* __restrict__ kvpts,
                         const float* __restrict__ pb,  const float* __restrict__ mask,
                         const float* __restrict__ head_w, float* __restrict__ Aout)
{
  const int h = blockIdx.x % NH;
  const int i = blockIdx.x / NH;
  __shared__ float lrow[NRES];
  __shared__ float rbuf[256];
  __shared__ float qs[CH];
  __shared__ float qp[PQ * 3];
  const int t = threadIdx.x;
  if (t < CH)     qs[t] = q[(size_t)i * (NH*CH) + h*CH + t];
  if (t < PQ*3)   qp[t] = qpts[(size_t)i * (NH*PQ*3) + h*(PQ*3) + t];
  __syncthreads();

  float sqq = 0.f;
#pragma unroll
  for (int x = 0; x < PQ*3; ++x) sqq += qp[x] * qp[x];
  // softplus(head_w) * sqrt(1/(3*PQ*4.5)) ; sqrt(1/54)=0.13608276
  const float hw = log1pf(expf(head_w[h])) * 0.13608276348795434f;
  const float mi = mask[i];

  float lmax = -1e30f;
  for (int j = t; j < NRES; j += 256) {
    // k slice: 16 floats, 16B aligned (j*384 + h*32) -> 4x b128
    const v4f* krow4 = (const v4f*)(kv + (size_t)j * (2*NH*CH) + h * (2*CH));
    float d = 0.f;
#pragma unroll
    for (int c4 = 0; c4 < 4; ++c4) {
      v4f kvv = krow4[c4];
      d += qs[c4*4+0]*kvv.x + qs[c4*4+1]*kvv.y + qs[c4*4+2]*kvv.z + qs[c4*4+3]*kvv.w;
    }
    float aa = (d * 0.25f + pb[((size_t)i*NRES + j) * NH + h]) * 0.5773502691896258f;
    // k_pts slice: 12 floats, 8B aligned (j*432 + h*36) -> 6x b64
    const v2f* kp2 = (const v2f*)(kvpts + (size_t)j * (NH*(PQ+PV)*3) + h * ((PQ+PV)*3));
    float sqk = 0.f, dot = 0.f;
#pragma unroll
    for (int x2 = 0; x2 < 6; ++x2) {
      v2f kx = kp2[x2];
      sqk += kx.x*kx.x + kx.y*kx.y;
      dot += qp[x2*2]*kx.x + qp[x2*2+1]*kx.y;
    }
    float pt  = (sqq + sqk - 2.f*dot) * hw * -0.5f;
    float msk = (mi * mask[j] - 1.f) * 100000.f;
    float l = aa + pt + msk;
    lrow[j] = l;
    lmax = fmaxf(lmax, l);
  }
  rbuf[t] = lmax; __syncthreads();
  for (int s = 128; s > 0; s >>= 1) { if (t < s) rbuf[t] = fmaxf(rbuf[t], rbuf[t+s]); __syncthreads(); }
  const float gmax = rbuf[0];
  __syncthreads();

  float lsum = 0.f;
  for (int j = t; j < NRES; j += 256) { float e = expf(lrow[j] - gmax); lrow[j] = e; lsum += e; }
  rbuf[t] = lsum; __syncthreads();
  for (int s = 128; s > 0; s >>= 1) { if (t < s) rbuf[t] += rbuf[t+s]; __syncthreads(); }
  const float inv = 1.f / rbuf[0];
  for (int j = t; j < NRES; j += 256)
    Aout[(size_t)h*NRES*NRES + (size_t)i*NRES + j] = lrow[j] * inv;
}

// ---------------------------------------------------------------------------
// o_pt -> local frame (rot^T * (o_pt - trans)), write coords + norm into feats
// ---------------------------------------------------------------------------
__global__ void finalize_opt(const float* __restrict__ optg, const float* __restrict__ rot,
                             const float* __restrict__ trans, float* __restrict__ feats)
{
  int idx = blockIdx.x * blockDim.x + threadIdx.x;
  if (idx >= NRES * NH * PV) return;
  int n  = idx / (NH*PV);
  int hp = idx % (NH*PV);
  const float* g = optg + (size_t)n * (NH*PV*3) + hp * 3;
  float dx = g[0] - trans[n*3+0];
  float dy = g[1] - trans[n*3+1];
  float dz = g[2] - trans[n*3+2];
  const float* R = rot + (size_t)n * 9;
  float lx = R[0]*dx + R[3]*dy + R[6]*dz;   // rot^T
  float ly = R[1]*dx + R[4]*dy + R[7]*dz;
  float lz = R[2]*dx + R[5]*dy + R[8]*dz;
  float* f = feats + (size_t)n * 2208;
  f[192 + hp*3 + 0] = lx;
  f[192 + hp*3 + 1] = ly;
  f[192 + hp*3 + 2] = lz;
  f[480 + hp] = sqrtf(lx*lx + ly*ly + lz*lz + 1e-8f);
}

// ---------------------------------------------------------------------------
// out = LayerNorm(x + y) ; one 128-thread block per row of 384
// ---------------------------------------------------------------------------
__global__ __launch_bounds__(128)
void add_layernorm(const float* __restrict__ x, const float* __restrict__ y,
                   const float* __restrict__ g, const float* __restrict__ b,
                   float* __restrict__ out)
{
  const int row = blockIdx.x;
  const int t = threadIdx.x;
  __shared__ float r1[128], r2[128];
  float vals[3]; float s = 0.f, s2 = 0.f;
#pragma unroll
  for (int k = 0; k < 3; ++k) {
    int c = t + k*128;
    float v = x[(size_t)row*CS + c] + y[(size_t)row*CS + c];
    vals[k] = v; s += v; s2 += v*v;
  }
  r1[t] = s; r2[t] = s2; __syncthreads();
  for (int st = 64; st > 0; st >>= 1) {
    if (t < st) { r1[t] += r1[t+st]; r2[t] += r2[t+st]; }
    __syncthreads();
  }
  const float mean = r1[0] * (1.f/CS);
  const float var  = r2[0] * (1.f/CS) - mean*mean;
  const float rstd = rsqrtf(var + 1e-5f);
#pragma unroll
  for (int k = 0; k < 3; ++k) {
    int c = t + k*128;
    out[(size_t)row*CS + c] = (vals[k] - mean) * rstd * g[c] + b[c];
  }
}

// ---------------------------------------------------------------------------
static inline void launch_gemm(hipStream_t st,
                               const float* A, int lda, long Astride,
                               const float* B, int ldb, long Bstride,
                               const float* bias,
                               float* C, int ldc, long Cstride,
                               int M, int K, int Nc, int batch, int relu)
{
  int tiles = ((M + 15) / 16) * ((Nc + 15) / 16);
  dim3 grid((tiles + 7) / 8, batch);
  gemm_f32_wmma<<<grid, 256, 0, st>>>(A, lda, Astride, B, ldb, Bstride,
                                      bias, C, ldc, Cstride, M, K, Nc, relu);
}

extern "C" void kernel_launch(void* const* d_in, const int* in_sizes, int n_in,
                              void* d_out, int out_size, void* d_ws, size_t ws_size,
                              hipStream_t stream)
{
  const float* s       = (const float*)d_in[0];
  const float* z       = (const float*)d_in[1];
  const float* rot     = (const float*)d_in[2];
  const float* trans   = (const float*)d_in[3];
  const float* mask    = (const float*)d_in[4];
  const float* wq      = (const float*)d_in[5];
  const float* bq      = (const float*)d_in[6];
  const float* wkv     = (const float*)d_in[7];
  const float* bkv     = (const float*)d_in[8];
  const float* wq_pts  = (const float*)d_in[9];
  const float* bq_pts  = (const float*)d_in[10];
  const float* bwkv_pts= (const float*)d_in[11];
  const float* bkv_pts = (const float*)d_in[12];
  const float* wb      = (const float*)d_in[13];
  const float* bbias   = (const float*)d_in[14];
  const float* head_w  = (const float*)d_in[15];
  const float* wo      = (const float*)d_in[16];
  const float* bo      = (const float*)d_in[17];
  const float* ln1_g   = (const float*)d_in[18];
  const float* ln1_b   = (const float*)d_in[19];
  const float* wt1     = (const float*)d_in[20];
  const float* bt1     = (const float*)d_in[21];
  const float* wt2     = (const float*)d_in[22];
  const float* bt2     = (const float*)d_in[23];
  const float* wt3     = (const float*)d_in[24];
  const float* bt3     = (const float*)d_in[25];
  const float* ln2_g   = (const float*)d_in[26];
  const float* ln2_b   = (const float*)d_in[27];
  const float* w_bb    = (const float*)d_in[28];
  const float* b_bb    = (const float*)d_in[29];

  // workspace carve-up (floats); ~75 MB total
  float* ws     = (float*)d_ws;
  float* q      = ws;                              // 768*192
  float* kvb    = q       + (size_t)NRES*192;      // 768*384 (k|v interleaved)
  float* qp_lin = kvb     + (size_t)NRES*384;      // 768*144
  float* q_pts  = qp_lin  + (size_t)NRES*144;      // 768*144
  float* kvp_lin= q_pts   + (size_t)NRES*144;      // 768*432
  float* kv_pts = kvp_lin + (size_t)NRES*432;      // 768*432
  float* pb     = kv_pts  + (size_t)NRES*432;      // 768*768*12
  float* Abuf   = pb      + (size_t)NRES*NRES*NH;  // 12*768*768
  float* optg   = Abuf    + (size_t)NH*NRES*NRES;  // 768*288
  float* feats  = optg    + (size_t)NRES*NH*PV*3;  // 768*2208
  float* s_upd  = feats   + (size_t)NRES*2208;     // 768*384
  float* s1     = s_upd   + (size_t)NRES*CS;       // 768*384
  float* t1     = s1      + (size_t)NRES*CS;       // 768*384
  float* t2     = t1      + (size_t)NRES*CS;       // 768*384

  // 1. input projections from s
  launch_gemm(stream, s, CS, 0, wq,       192, 0, bq,      q,       192, 0, NRES, CS, 192, 1, 0);
  launch_gemm(stream, s, CS, 0, wkv,      384, 0, bkv,     kvb,     384, 0, NRES, CS, 384, 1, 0);
  launch_gemm(stream, s, CS, 0, wq_pts,   144, 0, bq_pts,  qp_lin,  144, 0, NRES, CS, 144, 1, 0);
  launch_gemm(stream, s, CS, 0, bwkv_pts, 432, 0, bkv_pts, kvp_lin, 432, 0, NRES, CS, 432, 1, 0);

  // 2. rigid transform to global frame
  { int tot = NRES*48;  pts_to_global<<<(tot+255)/256,256,0,stream>>>(qp_lin,  rot, trans, q_pts,  48); }
  { int tot = NRES*144; pts_to_global<<<(tot+255)/256,256,0,stream>>>(kvp_lin, rot, trans, kv_pts, 144); }

  // 3. pair bias: z (N*N x 128) @ wb (128 x 12)   -- first z pass
  launch_gemm(stream, z, CZ, 0, wb, NH, 0, bbias, pb, NH, 0, NRES*NRES, CZ, NH, 1, 0);

  // 4. attention logits + softmax -> Abuf[h, i, j]
  attn_logits_softmax<<<NH*NRES, 256, 0, stream>>>(q, kvb, q_pts, kv_pts, pb, mask, head_w, Abuf);

  // 5a. o = A_h @ v_h  -> feats[:, 0:192]           (batch over heads)
  launch_gemm(stream, Abuf, NRES, (long)NRES*NRES,
              kvb + CH, 2*NH*CH, 2*CH, nullptr,
              feats, 2208, CH, NRES, NRES, CH, NH, 0);
  // 5b. o_pt = A_h @ v_pts_h -> optg                (batch over heads)
  launch_gemm(stream, Abuf, NRES, (long)NRES*NRES,
              kv_pts + PQ*3, NH*(PQ+PV)*3, (PQ+PV)*3, nullptr,
              optg, NH*PV*3, PV*3, NRES, NRES, PV*3, NH, 0);
  // 5c. o_pair = A_i (12x768) @ z_i (768x128) -> feats[:, 576:2112]  -- second z pass
  launch_gemm(stream, Abuf, (int)(NRES*NRES), (long)NRES,
              z, CZ, (long)NRES*CZ, nullptr,
              feats + 576, CZ, 2208, NH, NRES, CZ, NRES, 0);

  // 6. o_pt -> local frame + norms into feats[:, 192:480] and [480:576]
  { int tot = NRES*NH*PV; finalize_opt<<<(tot+255)/256,256,0,stream>>>(optg, rot, trans, feats); }

  // 7. s_upd = feats @ wo + bo ; s1 = LN(s + s_upd)
  launch_gemm(stream, feats, 2208, 0, wo, CS, 0, bo, s_upd, CS, 0, NRES, 2208, CS, 1, 0);
  add_layernorm<<<NRES, 128, 0, stream>>>(s, s_upd, ln1_g, ln1_b, s1);

  // 8. transition MLP (relu fused) ; s2 = LN(s1 + t3) -> d_out
  launch_gemm(stream, s1, CS, 0, wt1, CS, 0, bt1, t1,    CS, 0, NRES, CS, CS, 1, 1);
  launch_gemm(stream, t1, CS, 0, wt2, CS, 0, bt2, t2,    CS, 0, NRES, CS, CS, 1, 1);
  launch_gemm(stream, t2, CS, 0, wt3, CS, 0, bt3, s_upd, CS, 0, NRES, CS, CS, 1, 0); // t3 reuses s_upd
  float* out = (float*)d_out;
  add_layernorm<<<NRES, 128, 0, stream>>>(s1, s_upd, ln2_g, ln2_b, out);             // s2

  // 9. bb_upd = s2 @ w_bb + b_bb -> d_out[294912:]
  launch_gemm(stream, out, CS, 0, w_bb, 6, 0, b_bb, out + (size_t)NRES*CS, 6, 0,
              NRES, CS, 6, 1, 0);
}